// ReconGNN_26173530702629
// MI455X (gfx1250) — compile-verified
//
#include <hip/hip_runtime.h>
#include <stdint.h>

typedef __attribute__((ext_vector_type(2))) float v2f;
typedef __attribute__((ext_vector_type(8))) float v8f;

#define RG_NODE_IN 7
#define RG_EDGE_IN 8
#define RG_H 64
#define RG_HEADS 4
#define RG_DPH 16
#define RG_FIN (2 * RG_H + RG_EDGE_IN)   // 136
#define RG_NEG_SLOPE 0.2f

// ---------- order-preserving float<->uint mapping for atomic max ----------
__device__ __forceinline__ unsigned rg_ford(float f) {
    unsigned u = __float_as_uint(f);
    return (u & 0x80000000u) ? ~u : (u | 0x80000000u);
}
__device__ __forceinline__ float rg_forddec(unsigned u) {
    return __uint_as_float((u & 0x80000000u) ? (u & 0x7FFFFFFFu) : ~u);
}

// ---------- encoder: h = relu(x @ W_enc + b_enc), x is (N,7) ----------
__global__ void rg_encoder(const float* __restrict__ x, const float* __restrict__ W,
                           const float* __restrict__ b, float* __restrict__ h, int N) {
    int i = blockIdx.x * blockDim.x + threadIdx.x;
    if (i >= N * RG_H) return;
    int n = i >> 6, c = i & 63;
    float acc = b[c];
#pragma unroll
    for (int d = 0; d < RG_NODE_IN; ++d) acc += x[n * RG_NODE_IN + d] * W[d * RG_H + c];
    h[i] = fmaxf(acc, 0.f);
}

// ---------- dense 64x64 transform via V_WMMA_F32_16X16X4_F32 ----------
// out(N,64) = in(N,64) @ W(64,64) [+ bias] [relu]. One wave per 16-row tile.
__global__ void rg_gemm64_wmma(const float* __restrict__ A, const float* __restrict__ B,
                               const float* __restrict__ bias, float* __restrict__ out,
                               int N, int relu) {
    int lane = threadIdx.x;            // 0..31, one full wave32 (EXEC all 1s)
    int row0 = blockIdx.x * 16;
    int rlo  = lane & 15;
    int rsel = lane >> 4;              // 0: k/M lo half, 1: hi half
    v8f acc0 = {}, acc1 = {}, acc2 = {}, acc3 = {};

    for (int t = 0; t < 16; ++t) {     // K = 64 in steps of 4
        int k = 4 * t + (rsel << 1);
        int arow = row0 + rlo; if (arow >= N) arow = N - 1;   // clamp (no divergence)
        v2f a;
        a.x = A[arow * RG_H + k];
        a.y = A[arow * RG_H + k + 1];
#pragma unroll
        for (int c = 0; c < 4; ++c) {
            int col = c * 16 + rlo;
            v2f bb;
            bb.x = B[k * RG_H + col];
            bb.y = B[(k + 1) * RG_H + col];
            v8f* accp = (c == 0) ? &acc0 : (c == 1) ? &acc1 : (c == 2) ? &acc2 : &acc3;
            *accp = __builtin_amdgcn_wmma_f32_16x16x4_f32(
                /*neg_a=*/false, a, /*neg_b=*/false, bb,
                /*c_mod=*/(short)0, *accp, /*reuse_a=*/false, /*reuse_b=*/false);
        }
    }

#pragma unroll
    for (int c = 0; c < 4; ++c) {
        const v8f acc = (c == 0) ? acc0 : (c == 1) ? acc1 : (c == 2) ? acc2 : acc3;
        int col = c * 16 + rlo;
#pragma unroll
        for (int i = 0; i < 8; ++i) {
            int row = row0 + i + (rsel << 3);
            if (row < N) {
                float v = acc[i];
                if (bias) v += bias[col];
                if (relu) v = fmaxf(v, 0.f);
                out[row * RG_H + col] = v;
            }
        }
    }
}

// ---------- wae[d][h] = sum_k We[d, h*16+k] * a_e[h*16+k]  (8x4) ----------
__global__ void rg_wae(const float* __restrict__ We, const float* __restrict__ ae,
                       float* __restrict__ wae) {
    int i = threadIdx.x;
    if (i >= RG_EDGE_IN * RG_HEADS) return;
    int d = i >> 2, hd = i & 3;
    float acc = 0.f;
#pragma unroll
    for (int k = 0; k < RG_DPH; ++k) acc += We[d * RG_H + hd * RG_DPH + k] * ae[hd * RG_DPH + k];
    wae[i] = acc;
}

// ---------- al_s / al_d from g ----------
__global__ void rg_alpha_node(const float* __restrict__ g, const float* __restrict__ asrc,
                              const float* __restrict__ adst, float* __restrict__ al_s,
                              float* __restrict__ al_d, int N) {
    int i = blockIdx.x * blockDim.x + threadIdx.x;
    if (i >= N * RG_HEADS) return;
    int n = i >> 2, hd = i & 3;
    const float* gp = g + n * RG_H + hd * RG_DPH;
    float s = 0.f, dv = 0.f;
#pragma unroll
    for (int k = 0; k < RG_DPH; ++k) {
        float gv = gp[k];
        s  += gv * asrc[hd * RG_DPH + k];
        dv += gv * adst[hd * RG_DPH + k];
    }
    al_s[i] = s;
    al_d[i] = dv;
}

// ---------- per-layer zero init: m (ordered -inf == 0), den, agg ----------
__global__ void rg_init_layer(unsigned* __restrict__ m, float* __restrict__ den,
                              float* __restrict__ agg, int N) {
    int i = blockIdx.x * blockDim.x + threadIdx.x;
    if (i < N * RG_HEADS) { m[i] = 0u; den[i] = 0.f; }
    if (i < N * RG_H) agg[i] = 0.f;
}

// ---------- edge logits + leaky relu + segment max ----------
__global__ void rg_edge_logits(const int* __restrict__ src, const int* __restrict__ dst,
                               const float* __restrict__ ea, const float* __restrict__ al_s,
                               const float* __restrict__ al_d, const float* __restrict__ wae,
                               float* __restrict__ ebuf, unsigned* __restrict__ m, int E) {
    int i = blockIdx.x * blockDim.x + threadIdx.x;
    if (i >= E * RG_HEADS) return;
    int e = i >> 2, hd = i & 3;
    float ale = 0.f;
#pragma unroll
    for (int d = 0; d < RG_EDGE_IN; ++d) ale += ea[e * RG_EDGE_IN + d] * wae[d * RG_HEADS + hd];
    int s = src[e], t = dst[e];
    float v = al_s[s * RG_HEADS + hd] + al_d[t * RG_HEADS + hd] + ale;
    v = (v > 0.f) ? v : RG_NEG_SLOPE * v;
    ebuf[i] = v;
    atomicMax(&m[t * RG_HEADS + hd], rg_ford(v));
}

// ---------- exp(e - m[dst]) and segment denominator ----------
__global__ void rg_edge_exp(const int* __restrict__ dst, float* __restrict__ ebuf,
                            const unsigned* __restrict__ m, float* __restrict__ den, int E) {
    int i = blockIdx.x * blockDim.x + threadIdx.x;
    if (i >= E * RG_HEADS) return;
    int e = i >> 2, hd = i & 3;
    int t = dst[e];
    float ex = __expf(ebuf[i] - rg_forddec(m[t * RG_HEADS + hd]));
    ebuf[i] = ex;
    atomicAdd(&den[t * RG_HEADS + hd], ex);
}

// ---------- attention-weighted scatter: agg[dst] += attn * g[src] ----------
__global__ void rg_edge_aggregate(const int* __restrict__ src, const int* __restrict__ dst,
                                  const float* __restrict__ ebuf, const float* __restrict__ den,
                                  const float* __restrict__ g, float* __restrict__ agg, int E) {
    int i = blockIdx.x * blockDim.x + threadIdx.x;
    if (i >= E * RG_HEADS) return;
    int e = i >> 2, hd = i & 3;
    int s = src[e], t = dst[e];
    float a = ebuf[i] / fmaxf(den[t * RG_HEADS + hd], 1e-16f);
    const float* gp = g + s * RG_H + hd * RG_DPH;
    float* op = agg + t * RG_H + hd * RG_DPH;
#pragma unroll
    for (int k = 0; k < RG_DPH; ++k) atomicAdd(&op[k], a * gp[k]);
}

// ---------- h = relu(agg + b) ----------
__global__ void rg_bias_relu(const float* __restrict__ agg, const float* __restrict__ b,
                             float* __restrict__ h, int N) {
    int i = blockIdx.x * blockDim.x + threadIdx.x;
    if (i >= N * RG_H) return;
    h[i] = fmaxf(agg[i] + b[i & 63], 0.f);
}

// ---------- pressure head layer 2: p = hid @ Wp2 + bp2 ----------
__global__ void rg_pressure_out(const float* __restrict__ hid, const float* __restrict__ Wp2,
                                const float* __restrict__ bp2, float* __restrict__ p, int N) {
    int n = blockIdx.x * blockDim.x + threadIdx.x;
    if (n >= N) return;
    float acc = bp2[0];
#pragma unroll 8
    for (int c = 0; c < RG_H; ++c) acc += hid[n * RG_H + c] * Wp2[c];
    p[n] = acc;
}

// ---------- flow head via WMMA ----------
// hidden(16x64) = [h[src] | h[dst] | ea](16x136) @ Wf1(136x64); flow = relu(hidden+bf1) @ Wf2 + bf2
// One wave per 16-edge tile; Wf1 staged pair-interleaved in LDS, shared by 8 waves/block;
// each wave grid-strides over tiles. A-fragments gathered directly from global (lane L
// only ever feeds row L&15, k-pairs never straddle the 64/128 segment boundaries).
__global__ void rg_flow_wmma(const int* __restrict__ src, const int* __restrict__ dst,
                             const float* __restrict__ ea, const float* __restrict__ h,
                             const float* __restrict__ Wf1, const float* __restrict__ bf1,
                             const float* __restrict__ Wf2, const float* __restrict__ bf2,
                             float* __restrict__ flow, int EORIG, int nTiles) {
    __shared__ float lB[(RG_FIN / 2) * RG_H * 2];   // 68*64*2 floats = 34816 B

    // stage Wf1: pair p=(k/2) x col, (k, k+1) adjacent -> one ds_load_b64 per fragment
    for (int idx = threadIdx.x; idx < (RG_FIN / 2) * RG_H; idx += blockDim.x) {
        int p = idx >> 6, col = idx & 63;
        lB[idx * 2]     = Wf1[(2 * p) * RG_H + col];
        lB[idx * 2 + 1] = Wf1[(2 * p + 1) * RG_H + col];
    }
    __syncthreads();

    const int lane = threadIdx.x & 31;
    const int wave = threadIdx.x >> 5;
    const int wavesPerBlock = blockDim.x >> 5;
    const int rlo  = lane & 15;
    const int rsel = lane >> 4;
    const int totalWaves = gridDim.x * wavesPerBlock;

    for (int tile = blockIdx.x * wavesPerBlock + wave; tile < nTiles; tile += totalWaves) {
        int e0 = tile * 16;
        int er = e0 + rlo; if (er >= EORIG) er = EORIG - 1;   // clamp, no divergence
        const float* srow = h + (size_t)src[er] * RG_H;
        const float* drow = h + (size_t)dst[er] * RG_H;
        const float* erow = ea + (size_t)er * RG_EDGE_IN;

        // prefetch next tile's gather sources (global_prefetch_b8)
        int en = e0 + totalWaves * 16 + rlo;
        if (en < EORIG) {
            __builtin_prefetch(&src[en], 0, 1);
            __builtin_prefetch(&dst[en], 0, 1);
            __builtin_prefetch(&ea[(size_t)en * RG_EDGE_IN], 0, 1);
        }

        v8f acc0 = {}, acc1 = {}, acc2 = {}, acc3 = {};
        for (int t = 0; t < RG_FIN / 4; ++t) {               // 34 K-steps
            int kk = 4 * t + (rsel << 1);
            // branchless segment select; both half-waves always in same segment
            const float* ap = (kk < RG_H)       ? (srow + kk)
                            : (kk < 2 * RG_H)   ? (drow + (kk - RG_H))
                                                : (erow + (kk - 2 * RG_H));
            v2f a = *(const v2f*)ap;                         // 8B-aligned (kk even)
            int pbase = (kk >> 1) * RG_H;                    // pair index * 64
#pragma unroll
            for (int c = 0; c < 4; ++c) {
                v2f bb = *(const v2f*)&lB[(pbase + c * 16 + rlo) * 2];
                v8f* accp = (c == 0) ? &acc0 : (c == 1) ? &acc1 : (c == 2) ? &acc2 : &acc3;
                *accp = __builtin_amdgcn_wmma_f32_16x16x4_f32(
                    false, a, false, bb, (short)0, *accp, false, false);
            }
        }

        // epilogue: bias + relu + dot with Wf2, reduce 16 lanes per half-wave
        float partial[8];
#pragma unroll
        for (int i = 0; i < 8; ++i) partial[i] = 0.f;
#pragma unroll
        for (int c = 0; c < 4; ++c) {
            const v8f acc = (c == 0) ? acc0 : (c == 1) ? acc1 : (c == 2) ? acc2 : acc3;
            int col = c * 16 + rlo;
            float b1 = bf1[col], w2 = Wf2[col];
#pragma unroll
            for (int i = 0; i < 8; ++i)
                partial[i] += fmaxf(acc[i] + b1, 0.f) * w2;
        }
#pragma unroll
        for (int off = 1; off < 16; off <<= 1) {
#pragma unroll
            for (int i = 0; i < 8; ++i)
                partial[i] += __shfl_xor(partial[i], off, 32);
        }
        if (rlo == 0) {
            float b2 = bf2[0];
#pragma unroll
            for (int i = 0; i < 8; ++i) {
                int e = e0 + i + (rsel << 3);               // D rows: M=i / M=i+8
                if (e < EORIG) flow[e] = partial[i] + b2;
            }
        }
    }
}

extern "C" void kernel_launch(void* const* d_in, const int* in_sizes, int n_in,
                              void* d_out, int out_size, void* d_ws, size_t ws_size,
                              hipStream_t stream) {
    // ---- inputs (setup_inputs dict order) ----
    const float* x     = (const float*)d_in[0];
    const int*   ei    = (const int*)  d_in[1];
    const float* ea    = (const float*)d_in[2];
    // d_in[3] is_original_edge: originals are edges [0, E_ORIG) by construction
    const float* W_enc = (const float*)d_in[4];
    const float* b_enc = (const float*)d_in[5];
    const float* Wl[2]    = {(const float*)d_in[6],  (const float*)d_in[12]};
    const float* Wel[2]   = {(const float*)d_in[7],  (const float*)d_in[13]};
    const float* asrcl[2] = {(const float*)d_in[8],  (const float*)d_in[14]};
    const float* adstl[2] = {(const float*)d_in[9],  (const float*)d_in[15]};
    const float* ael[2]   = {(const float*)d_in[10], (const float*)d_in[16]};
    const float* bl[2]    = {(const float*)d_in[11], (const float*)d_in[17]};
    const float* Wp1 = (const float*)d_in[18];
    const float* bp1 = (const float*)d_in[19];
    const float* Wp2 = (const float*)d_in[20];
    const float* bp2 = (const float*)d_in[21];
    const float* Wf1 = (const float*)d_in[22];
    const float* bf1 = (const float*)d_in[23];
    const float* Wf2 = (const float*)d_in[24];
    const float* bf2 = (const float*)d_in[25];

    const int N     = in_sizes[0] / RG_NODE_IN;
    const int E     = in_sizes[1] / 2;
    const int EORIG = E / 2;
    const int* src = ei;
    const int* dst = ei + E;

    // ---- outputs: [pressure(N) | flow(EORIG) | h(N*64)] ----
    float* p    = (float*)d_out;
    float* flow = p + N;
    float* h    = flow + EORIG;   // final h lives here; used as working h buffer

    // ---- workspace ----
    float*    g    = (float*)d_ws;                         // N*64
    float*    agg  = g    + (size_t)N * RG_H;              // N*64
    float*    ebuf = agg  + (size_t)N * RG_H;              // E*4
    float*    al_s = ebuf + (size_t)E * RG_HEADS;          // N*4
    float*    al_d = al_s + (size_t)N * RG_HEADS;          // N*4
    unsigned* m    = (unsigned*)(al_d + (size_t)N * RG_HEADS); // N*4
    float*    den  = (float*)(m + (size_t)N * RG_HEADS);   // N*4
    float*    wae  = den + (size_t)N * RG_HEADS;           // 32

    const int B256 = 256;
    const int gNH  = (N * RG_H + B256 - 1) / B256;
    const int gN4  = (N * RG_HEADS + B256 - 1) / B256;
    const int gE4  = (E * RG_HEADS + B256 - 1) / B256;
    const int gTiles = (N + 15) / 16;

    // encoder
    rg_encoder<<<gNH, B256, 0, stream>>>(x, W_enc, b_enc, h, N);

    // two GAT layers
    for (int L = 0; L < 2; ++L) {
        rg_init_layer<<<gNH, B256, 0, stream>>>(m, den, agg, N);
        rg_gemm64_wmma<<<gTiles, 32, 0, stream>>>(h, Wl[L], nullptr, g, N, 0);
        rg_alpha_node<<<gN4, B256, 0, stream>>>(g, asrcl[L], adstl[L], al_s, al_d, N);
        rg_wae<<<1, 32, 0, stream>>>(Wel[L], ael[L], wae);
        rg_edge_logits<<<gE4, B256, 0, stream>>>(src, dst, ea, al_s, al_d, wae, ebuf, m, E);
        rg_edge_exp<<<gE4, B256, 0, stream>>>(dst, ebuf, m, den, E);
        rg_edge_aggregate<<<gE4, B256, 0, stream>>>(src, dst, ebuf, den, g, agg, E);
        rg_bias_relu<<<gNH, B256, 0, stream>>>(agg, bl[L], h, N);
    }

    // pressure head: hidden = relu(h@Wp1 + bp1) (WMMA, fused), then dot with Wp2
    rg_gemm64_wmma<<<gTiles, 32, 0, stream>>>(h, Wp1, bp1, g, N, 1);
    rg_pressure_out<<<(N + B256 - 1) / B256, B256, 0, stream>>>(g, Wp2, bp2, p, N);

    // flow head over the original EORIG edges (WMMA, 16 edges per wave-tile)
    const int nTiles = (EORIG + 15) / 16;
    int flowBlocks = (nTiles + 7) / 8;
    if (flowBlocks > 1024) flowBlocks = 1024;
    rg_flow_wmma<<<flowBlocks, 256, 0, stream>>>(src, dst, ea, h, Wf1, bf1, Wf2, bf2,
                                                 flow, EORIG, nTiles);
}